// GCN_76201309766176
// MI455X (gfx1250) — compile-verified
//
#include <hip/hip_runtime.h>
#include <hip/hip_bf16.h>
#include <stdint.h>

// ---------------------------------------------------------------------------
// GCN (DGL GraphConv, norm='both'):
//   out[d] = in_norm[d] * sum_{edges (s,d)} out_norm[s] * feat[s]
// Memory-bound scatter/gather (~0.25 FLOP/byte) -> no dense GEMM, no WMMA.
// All working sets are L2-resident on MI455X (192 MB): optimize for L2
// gather + non-returning f32 atomic throughput, use gfx1250 async-to-LDS
// for edge-index staging and global_prefetch_b8 for the random gathers.
// ---------------------------------------------------------------------------

#define FEAT 32

#if defined(__AMDGCN__) && __has_builtin(__builtin_amdgcn_global_load_async_to_lds_b32)
#define GCN_ASYNC_LDS 1
typedef __attribute__((address_space(1))) int gcn_gint;   // global (AS1) int
typedef __attribute__((address_space(3))) int gcn_lint;   // LDS    (AS3) int
#endif

// Zero output accumulator and degree counters (harness poisons d_out/d_ws).
__global__ void gcn_init_kernel(float* __restrict__ out, unsigned* __restrict__ odeg,
                                unsigned* __restrict__ ideg, int n_out, int n_nodes) {
    int i = blockIdx.x * blockDim.x + threadIdx.x;
    int stride = gridDim.x * blockDim.x;
    for (int k = i; k < n_out; k += stride) out[k] = 0.0f;
    for (int k = i; k < n_nodes; k += stride) { odeg[k] = 0u; ideg[k] = 0u; }
}

// Per-edge degree counting (u32 atomics, non-returning).
__global__ void gcn_degree_kernel(const int* __restrict__ src, const int* __restrict__ dst,
                                  unsigned* __restrict__ odeg, unsigned* __restrict__ ideg,
                                  int n_edges) {
    int i = blockIdx.x * blockDim.x + threadIdx.x;
    int stride = gridDim.x * blockDim.x;
    for (int e = i; e < n_edges; e += stride) {
        atomicAdd(&odeg[src[e]], 1u);
        atomicAdd(&ideg[dst[e]], 1u);
    }
}

// norm = 1/sqrt(max(deg,1)); tiny kernel, accurate sqrt path.
__global__ void gcn_norm_kernel(const unsigned* __restrict__ odeg,
                                const unsigned* __restrict__ ideg,
                                float* __restrict__ onorm, float* __restrict__ inorm,
                                int n_nodes) {
    int i = blockIdx.x * blockDim.x + threadIdx.x;
    int stride = gridDim.x * blockDim.x;
    for (int k = i; k < n_nodes; k += stride) {
        unsigned od = odeg[k]; if (od < 1u) od = 1u;
        unsigned id = ideg[k]; if (id < 1u) id = 1u;
        onorm[k] = 1.0f / sqrtf((float)od);
        inorm[k] = 1.0f / sqrtf((float)id);
    }
}

// h = feat * out_norm (source-side prescale; removes norm work from hot loop).
__global__ void gcn_prescale_kernel(const float* __restrict__ feat,
                                    const float* __restrict__ onorm,
                                    float* __restrict__ h, int n_out) {
    int i = blockIdx.x * blockDim.x + threadIdx.x;
    int stride = gridDim.x * blockDim.x;
    for (int k = i; k < n_out; k += stride) h[k] = feat[k] * onorm[k >> 5];
}

static __device__ __forceinline__ float bcast_f32(float v, int lane) {
    return __uint_as_float(__builtin_amdgcn_readlane(__float_as_uint(v), lane));
}

// Core scatter. One wave32 per 32-edge chunk; lane = feature index.
//  - edge indices staged via gfx1250 async-to-LDS (ASYNCcnt) when available
//  - one global_prefetch_b8 prefetches all 32 scattered feature rows
//  - fully-unrolled inner loop: v_readlane broadcasts + 128B coalesced
//    gather + 128B coalesced non-returning global_atomic_add_f32
template <bool PRESCALED>
__global__ void gcn_scatter_kernel(const float* __restrict__ feat,
                                   const int* __restrict__ src,
                                   const int* __restrict__ dst,
                                   const float* __restrict__ onorm,
                                   float* __restrict__ out, int n_edges) {
    const int lane = threadIdx.x & 31;
    const int wave = (blockIdx.x * blockDim.x + threadIdx.x) >> 5;
    const int base = wave * 32;
    if (base >= n_edges) return;

    const int e  = base + lane;
    const int eg = (e < n_edges) ? e : (n_edges - 1);  // clamp: safe address,
                                                       // never used past cnt
    int s, d;
#ifdef GCN_ASYNC_LDS
    __shared__ int lds_idx[8 * 64];                 // 8 waves * (32 src + 32 dst)
    int* sslot = &lds_idx[(threadIdx.x >> 5) * 64];
    int* dslot = sslot + 32;
    // Per-lane async DMA: 32 src + 32 dst indices -> this wave's LDS slot.
    __builtin_amdgcn_global_load_async_to_lds_b32(
        (gcn_gint*)const_cast<int*>(src + eg),
        (gcn_lint*)(sslot + lane), 0, 0);
    __builtin_amdgcn_global_load_async_to_lds_b32(
        (gcn_gint*)const_cast<int*>(dst + eg),
        (gcn_lint*)(dslot + lane), 0, 0);
#if __has_builtin(__builtin_amdgcn_s_wait_asynccnt)
    __builtin_amdgcn_s_wait_asynccnt(0);
#else
    asm volatile("s_wait_asynccnt 0x0" ::: "memory");
#endif
    asm volatile("" ::: "memory");                  // don't hoist LDS reads
    s = sslot[lane];
    d = dslot[lane];
#else
    s = src[eg];
    d = dst[eg];
#endif
    float nrm = PRESCALED ? 1.0f : onorm[s];

    // One instruction: 32 lanes prefetch 32 scattered 128B feature rows.
    __builtin_prefetch((const void*)(feat + (size_t)s * FEAT), 0, 3);

    int cnt = n_edges - base;
    if (cnt >= 32) {
        // Fast path: full chunk, unrolled, immediate-lane v_readlane.
#pragma unroll
        for (int i = 0; i < 32; ++i) {
            int   ss = __builtin_amdgcn_readlane(s, i);
            int   dd = __builtin_amdgcn_readlane(d, i);
            float v  = feat[ss * FEAT + lane];
            if (!PRESCALED) v *= bcast_f32(nrm, i);
            atomicAdd(&out[dd * FEAT + lane], v);
        }
    } else {
        for (int i = 0; i < cnt; ++i) {             // tail chunk (uniform i)
            int   ss = __builtin_amdgcn_readlane(s, i);
            int   dd = __builtin_amdgcn_readlane(d, i);
            float v  = feat[ss * FEAT + lane];
            if (!PRESCALED) v *= bcast_f32(nrm, i);
            atomicAdd(&out[dd * FEAT + lane], v);
        }
    }
}

// Final per-destination normalization: out[node, f] *= in_norm[node].
__global__ void gcn_scale_kernel(float* __restrict__ out, const float* __restrict__ inorm,
                                 int n_out) {
    int i = blockIdx.x * blockDim.x + threadIdx.x;
    int stride = gridDim.x * blockDim.x;
    for (int k = i; k < n_out; k += stride) out[k] *= inorm[k >> 5];
}

extern "C" void kernel_launch(void* const* d_in, const int* in_sizes, int n_in,
                              void* d_out, int out_size, void* d_ws, size_t ws_size,
                              hipStream_t stream) {
    const float* feat = (const float*)d_in[0];
    const int*   src  = (const int*)d_in[1];
    const int*   dst  = (const int*)d_in[2];
    float*       out  = (float*)d_out;

    const int n_nodes = in_sizes[0] / FEAT;   // 100000
    const int n_edges = in_sizes[1];          // 1600000
    const int n_out   = out_size;             // n_nodes * FEAT
    (void)n_in;

    // Workspace layout: [odeg u32 | ideg u32 | onorm f32 | inorm f32 | h f32*FEAT]
    unsigned* odeg  = (unsigned*)d_ws;
    unsigned* ideg  = odeg + n_nodes;
    float*    onorm = (float*)(ideg + n_nodes);
    float*    inorm = onorm + n_nodes;
    float*    h     = inorm + n_nodes;
    const size_t need_base = (size_t)n_nodes * 16u;
    const size_t need_h    = need_base + (size_t)n_out * 4u;
    const bool   prescale  = (ws_size >= need_h);

    const int T = 256;  // 8 wave32s per block

    {   // 1) zero accumulators
        int work   = n_out > n_nodes ? n_out : n_nodes;
        int blocks = (work + T - 1) / T;
        gcn_init_kernel<<<blocks, T, 0, stream>>>(out, odeg, ideg, n_out, n_nodes);
    }
    {   // 2) degree counting
        int blocks = (n_edges + T - 1) / T;
        gcn_degree_kernel<<<blocks, T, 0, stream>>>(src, dst, odeg, ideg, n_edges);
    }
    {   // 3) norms
        int blocks = (n_nodes + T - 1) / T;
        gcn_norm_kernel<<<blocks, T, 0, stream>>>(odeg, ideg, onorm, inorm, n_nodes);
    }
    if (prescale) {
        // 3b) h = feat * out_norm
        int blocks = (n_out + T - 1) / T;
        gcn_prescale_kernel<<<blocks, T, 0, stream>>>(feat, onorm, h, n_out);
    }
    {   // 4) edge scatter: one wave per 32-edge chunk
        int waves  = (n_edges + 31) / 32;
        int blocks = (waves * 32 + T - 1) / T;
        if (blocks < 1) blocks = 1;
        if (prescale)
            gcn_scatter_kernel<true><<<blocks, T, 0, stream>>>(h, src, dst, nullptr, out, n_edges);
        else
            gcn_scatter_kernel<false><<<blocks, T, 0, stream>>>(feat, src, dst, onorm, out, n_edges);
    }
    {   // 5) destination-side normalization
        int blocks = (n_out + T - 1) / T;
        gcn_scale_kernel<<<blocks, T, 0, stream>>>(out, inorm, n_out);
    }
}